// PairwiseMamba_16956530884945
// MI455X (gfx1250) — compile-verified
//
#include <hip/hip_runtime.h>
#include <hip/hip_bf16.h>

typedef float v2f __attribute__((ext_vector_type(2)));
typedef float v8f __attribute__((ext_vector_type(8)));

#define T_LEN 2048
#define N_SEQ 4608          // 8*16*36
#define P_AGG 36
#define OUT_ELTS 2048       // 8*16*16
#define WAVES_PER_BLOCK 4
#define BLOCK (WAVES_PER_BLOCK * 32)
#define TC 64               // timesteps per LDS chunk
#define REC 32              // floats per per-timestep record
// record layout (floats): [0..3]=softplus(dt), [4..7]=dt*x, [8..11]=silu(z),
//                         [12..19]=B[s], [20..27]=C[s], [28..31]=pad

static __device__ inline v8f wmma_f32_16x16x4(v2f a, v2f b, v8f c) {
    // 8-arg form: (neg_a, A, neg_b, B, c_mod, C, reuse_a, reuse_b)
    return __builtin_amdgcn_wmma_f32_16x16x4_f32(false, a, false, b,
                                                 (short)0, c, false, false);
}

// fast silu: x * rcp(1 + exp(-x))  -> v_exp_f32 + v_rcp_f32 (no IEEE div chain)
static __device__ inline float fast_silu(float x) {
    return x * __builtin_amdgcn_rcpf(1.0f + __expf(-x));
}
// fast softplus: log(1 + exp(x)) -> v_exp_f32 + v_log_f32 (args are ~0.1-scale)
static __device__ inline float fast_softplus(float x) {
    return __logf(1.0f + __expf(x));
}

__global__ void zero_out_kernel(float* __restrict__ out, int n) {
    int i = blockIdx.x * blockDim.x + threadIdx.x;
    if (i < n) out[i] = 0.0f;
}

__global__ __launch_bounds__(BLOCK)
void mamba_scan_kernel(const float* __restrict__ raw,   // [N_SEQ][2][T]
                       const float* __restrict__ ipw,   // [8][2]
                       const float* __restrict__ cw,    // [4][2]
                       const float* __restrict__ cb,    // [4]
                       const float* __restrict__ xpw,   // [17][4]
                       const float* __restrict__ dtw,   // [4][1]
                       const float* __restrict__ dtb,   // [4]
                       const float* __restrict__ Alog,  // [4][8]
                       const float* __restrict__ Dsk,   // [4]
                       const float* __restrict__ opw,   // [2][4]
                       const float* __restrict__ pw,    // [16][2]
                       const float* __restrict__ pb,    // [16]
                       float* __restrict__ out)         // [128][16]
{
    __shared__ float lds[WAVES_PER_BLOCK * TC * REC];   // 32 KB

    const int lane = threadIdx.x & 31;
    const int wave = threadIdx.x >> 5;
    const int n    = blockIdx.x * WAVES_PER_BLOCK + wave;
    float* rec = &lds[wave * TC * REC];

    const int d = lane >> 3;      // inner-channel index for scan phase
    const int s = lane & 7;       // state index for scan phase

    // ---- load tiny weight set into registers (uniform -> scalar loads) ----
    float w_ip[8][2];
#pragma unroll
    for (int r = 0; r < 8; ++r) { w_ip[r][0] = ipw[2*r]; w_ip[r][1] = ipw[2*r+1]; }
    float w_cw0[4], w_cw1[4], w_cb[4], w_dtrow[4], w_dtw[4], w_dtb[4], w_D[4];
#pragma unroll
    for (int r = 0; r < 4; ++r) {
        w_cw0[r] = cw[2*r]; w_cw1[r] = cw[2*r+1]; w_cb[r] = cb[r];
        w_dtrow[r] = xpw[r];            // x_proj row 0 (dt_rank==1)
        w_dtw[r] = dtw[r]; w_dtb[r] = dtb[r]; w_D[r] = Dsk[r];
    }
    float w_op[2][4];
#pragma unroll
    for (int c = 0; c < 2; ++c)
#pragma unroll
        for (int r = 0; r < 4; ++r) w_op[c][r] = opw[c*4 + r];

    // per-lane A = -exp(A_log[d][s])
    const float A_ls = -__expf(Alog[d*8 + s]);

    // WMMA B operand: x_proj rows 1..16 (B|C columns), ISA 4x16 layout
    const bool lo = lane < 16;
    const int  ll = lane & 15;
    v2f Bop;
    Bop.x = xpw[(1 + ll)*4 + (lo ? 0 : 2)];
    Bop.y = xpw[(1 + ll)*4 + (lo ? 1 : 3)];

    const float* u0p = raw + (size_t)n * 2 * T_LEN;   // channel 0 contiguous in t
    const float* u1p = u0p + T_LEN;                   // channel 1

    float h = 0.0f, acc = 0.0f;
    float skip[4] = {0.f, 0.f, 0.f, 0.f};

    for (int chunk = 0; chunk < T_LEN / TC; ++chunk) {
        const int t0 = chunk * TC;

        // ================= phase 1: time-parallel precompute =================
#pragma unroll
        for (int j = 0; j < TC / 32; ++j) {
            const int tl = j*32 + lane;
            const int tg = t0 + tl;
            float u0 = u0p[tg], u1 = u1p[tg];
            const int tgm = (tg > 0) ? tg - 1 : 0;
            float u0m = u0p[tgm], u1m = u1p[tgm];
            if (tg + TC < T_LEN) {                 // prefetch next chunk
                __builtin_prefetch(u0p + tg + TC, 0, 1);
                __builtin_prefetch(u1p + tg + TC, 0, 1);
            }

            float xs[4], sz[4], dt4[4], dtx[4];
            float dtraw = 0.0f;
#pragma unroll
            for (int dd = 0; dd < 4; ++dd) {
                float xl  = w_ip[dd][0]*u0  + w_ip[dd][1]*u1;       // in_proj x
                float zl  = w_ip[4+dd][0]*u0 + w_ip[4+dd][1]*u1;    // in_proj z
                float xlm = (tg > 0) ? (w_ip[dd][0]*u0m + w_ip[dd][1]*u1m) : 0.f;
                float xc  = xlm*w_cw0[dd] + xl*w_cw1[dd] + w_cb[dd]; // causal conv
                float xv  = fast_silu(xc);
                xs[dd] = xv;
                sz[dd] = fast_silu(zl);                              // silu(z)
                dtraw += w_dtrow[dd] * xv;
            }
#pragma unroll
            for (int dd = 0; dd < 4; ++dd) {
                float dl  = dtraw*w_dtw[dd] + w_dtb[dd];
                float dtv = fast_softplus(dl);
                dt4[dd] = dtv;
                dtx[dd] = dtv * xs[dd];
                skip[dd] += xs[dd] * w_D[dd] * sz[dd];               // D-skip term
            }
            float* rp = rec + tl*REC;
            *(float4*)(rp + 0) = make_float4(dt4[0], dt4[1], dt4[2], dt4[3]);
            *(float4*)(rp + 4) = make_float4(dtx[0], dtx[1], dtx[2], dtx[3]);
            *(float4*)(rp + 8) = make_float4(sz[0],  sz[1],  sz[2],  sz[3]);

            // ---- x_proj B|C columns via V_WMMA_F32_16X16X4_F32 ----
            // tile0: rows = t (lanes 0..15); tile1: rows = t (lanes 16..31)
            float sx2 = __shfl(xs[2], ll);        // K=2 of rows 0..15
            float sx3 = __shfl(xs[3], ll);        // K=3 of rows 0..15
            float sx0 = __shfl(xs[0], ll + 16);   // K=0 of rows 16..31
            float sx1 = __shfl(xs[1], ll + 16);   // K=1 of rows 16..31
            v2f A0, A1;
            A0.x = lo ? xs[0] : sx2;  A0.y = lo ? xs[1] : sx3;
            A1.x = lo ? sx0 : xs[2];  A1.y = lo ? sx1 : xs[3];
            v8f Z = {0.f,0.f,0.f,0.f,0.f,0.f,0.f,0.f};
            v8f D0 = wmma_f32_16x16x4(A0, Bop, Z);
            v8f D1 = wmma_f32_16x16x4(A1, Bop, Z);
            const int hi8 = lo ? 0 : 8;
#pragma unroll
            for (int r = 0; r < 8; ++r) {
                rec[(j*32      + r + hi8)*REC + 12 + ll] = D0[r];
                rec[(j*32 + 16 + r + hi8)*REC + 12 + ll] = D1[r];
            }
        }
        __syncthreads();

        // ================= phase 2: serial SSM scan over chunk ===============
#pragma unroll 8
        for (int t = 0; t < TC; ++t) {
            const float* r = rec + t*REC;
            float dtv  = r[d];
            float dtxv = r[4 + d];
            float szv  = r[8 + d];
            float Bv   = r[12 + s];
            float Cv   = r[20 + s];
            float dA = __expf(dtv * A_ls);
            h   = __builtin_fmaf(dA, h, dtxv * Bv);       // h = exp(dt*A)*h + dt*x*B
            acc = __builtin_fmaf(h, Cv * szv, acc);       // += h*C*silu(z)
        }
        __syncthreads();
    }

    // ---- final reductions ----
    // sum acc over s within each d-group of 8 lanes
#pragma unroll
    for (int o = 1; o < 8; o <<= 1) acc += __shfl_xor(acc, o);
    float ym[4];
    ym[0] = __shfl(acc, 0);  ym[1] = __shfl(acc, 8);
    ym[2] = __shfl(acc, 16); ym[3] = __shfl(acc, 24);
#pragma unroll
    for (int dd = 0; dd < 4; ++dd) {
        float sv = skip[dd];
#pragma unroll
        for (int o = 1; o < 32; o <<= 1) sv += __shfl_xor(sv, o);
        ym[dd] = (ym[dd] + sv) * (1.0f / (float)T_LEN);   // mean over t
    }
    float f0 = 0.f, f1 = 0.f;
#pragma unroll
    for (int dd = 0; dd < 4; ++dd) {
        f0 += w_op[0][dd] * ym[dd];
        f1 += w_op[1][dd] * ym[dd];
    }
    if (lane < 16) {
        float pv = f0 * pw[lane*2] + f1 * pw[lane*2 + 1] + pb[lane];
        pv = fmaxf(pv, 0.0f) * (1.0f / (float)P_AGG);     // relu + 1/P for mean
        atomicAdd(out + (n / P_AGG) * 16 + lane, pv);
    }
}

extern "C" void kernel_launch(void* const* d_in, const int* in_sizes, int n_in,
                              void* d_out, int out_size, void* d_ws, size_t ws_size,
                              hipStream_t stream) {
    const float* raw  = (const float*)d_in[0];
    const float* ipw  = (const float*)d_in[1];
    const float* cw   = (const float*)d_in[2];
    const float* cb   = (const float*)d_in[3];
    const float* xpw  = (const float*)d_in[4];
    const float* dtw  = (const float*)d_in[5];
    const float* dtb  = (const float*)d_in[6];
    const float* Alog = (const float*)d_in[7];
    const float* Dsk  = (const float*)d_in[8];
    const float* opw  = (const float*)d_in[9];
    const float* pw   = (const float*)d_in[10];
    const float* pb   = (const float*)d_in[11];
    float* out = (float*)d_out;

    zero_out_kernel<<<(OUT_ELTS + 255) / 256, 256, 0, stream>>>(out, OUT_ELTS);
    mamba_scan_kernel<<<N_SEQ / WAVES_PER_BLOCK, BLOCK, 0, stream>>>(
        raw, ipw, cw, cb, xpw, dtw, dtb, Alog, Dsk, opw, pw, pb, out);
}